// Linear_37830071943398
// MI455X (gfx1250) — compile-verified
//
#include <hip/hip_runtime.h>

typedef float v2f __attribute__((ext_vector_type(2)));
typedef float v8f __attribute__((ext_vector_type(8)));

#define N_TOK   2048
#define IN_F    2048
#define OUT_F   2048
#define BLOCK   32
#define OBLK    64
#define NNZ     1228
#define TOK_TILE 128
#define XS_STRIDE 36   // 144B rows: 16B aligned, conflict-free across 64 banks
#define WS_STRIDE 36
#define XBUF (TOK_TILE * XS_STRIDE)   // floats per X buffer
#define WBUF (BLOCK * WS_STRIDE)      // floats per W buffer

#if defined(__gfx1250__) && \
    __has_builtin(__builtin_amdgcn_global_load_async_to_lds_b128) && \
    __has_builtin(__builtin_amdgcn_s_wait_asynccnt)
#define USE_ASYNC 1
#else
#define USE_ASYNC 0
#endif

#if USE_ASYNC
typedef int v4i_b128 __attribute__((vector_size(16)));
typedef __attribute__((address_space(1))) v4i_b128* gas_v4i;   // global
typedef __attribute__((address_space(3))) v4i_b128* las_v4i;   // LDS
#endif

// Copy 16 bytes global -> LDS (async on CDNA5, sync fallback otherwise).
__device__ __forceinline__ void copy16_to_lds(const float* __restrict__ g,
                                              float* __restrict__ l) {
#if USE_ASYNC
    __builtin_amdgcn_global_load_async_to_lds_b128((gas_v4i)g, (las_v4i)l, 0, 0);
#else
    *(float4*)l = *(const float4*)g;
#endif
}

__device__ __forceinline__ void wait_stage_done() {
#if USE_ASYNC
    __builtin_amdgcn_s_wait_asynccnt(0);
#endif
}

// Stage one (X tile, W block) pair into an LDS buffer. 256 threads:
// W: 1024 floats = 1 x 16B per thread; X: 4096 floats = 4 x 16B per thread.
__device__ __forceinline__ void stage_tile(const float* __restrict__ x,
                                           const float* __restrict__ w,
                                           int tokBase, int ib, int k,
                                           float* __restrict__ XsBuf,
                                           float* __restrict__ WsBuf,
                                           int tid) {
    const float* xsrc = x + (size_t)tokBase * IN_F + (size_t)ib * BLOCK;
    const float* wsrc = w + (size_t)k * (BLOCK * BLOCK);
    {
        const int r  = tid >> 3;
        const int c4 = (tid & 7) << 2;
        copy16_to_lds(wsrc + r * BLOCK + c4, &WsBuf[r * WS_STRIDE + c4]);
    }
    #pragma unroll
    for (int j = 0; j < 4; ++j) {
        const int idx = tid + j * 256;
        const int r   = idx >> 3;
        const int c4  = (idx & 7) << 2;
        copy16_to_lds(xsrc + (size_t)r * IN_F + c4, &XsBuf[r * XS_STRIDE + c4]);
    }
}

// Kernel 1: segment starts via lower_bound on sorted oi.
__global__ void seg_start_kernel(const int* __restrict__ oi, int* __restrict__ seg) {
    int t = threadIdx.x;
    if (t <= OBLK) {
        int lo = 0, hi = NNZ;
        while (lo < hi) {
            int mid = (lo + hi) >> 1;
            if (oi[mid] < t) lo = mid + 1; else hi = mid;
        }
        seg[t] = lo;
    }
}

// Kernel 2: block-sparse GEMM, f32 WMMA, double-buffered async LDS staging.
__launch_bounds__(256)
__global__ void bsmm_wmma_kernel(const float* __restrict__ x,
                                 const float* __restrict__ w,
                                 const int*   __restrict__ ii,
                                 const int*   __restrict__ seg,
                                 float*       __restrict__ y) {
    __shared__ float Xs[2 * XBUF];
    __shared__ float Ws[2 * WBUF];

    const int ob      = blockIdx.x;            // output block (column group)
    const int tokBase = blockIdx.y * TOK_TILE; // first token of this tile
    const int tid     = threadIdx.x;
    const int lane    = tid & 31;
    const int wave    = tid >> 5;              // 0..7 -> 16-token strip
    const int rowM    = lane & 15;             // M (A/D) or N (B) index
    const int hiHalf  = lane >> 4;             // 0: K,K+1   1: K+2,K+3

    v8f acc0 = {}; // output cols ob*32 + [0,16)
    v8f acc1 = {}; // output cols ob*32 + [16,32)

    const int kBeg = seg[ob];
    const int kEnd = seg[ob + 1];

    // Prologue: stage first tile into buffer 0.
    if (kBeg < kEnd)
        stage_tile(x, w, tokBase, ii[kBeg], kBeg, &Xs[0], &Ws[0], tid);

    int p = 0;
    for (int k = kBeg; k < kEnd; ++k, p ^= 1) {
        wait_stage_done();   // this wave's async loads for tile k are in LDS
        __syncthreads();     // all waves: tile-k loads visible, buf[1-p] compute done

        if (k + 1 < kEnd)    // overlap next tile's loads with this tile's WMMAs
            stage_tile(x, w, tokBase, ii[k + 1], k + 1,
                       &Xs[(p ^ 1) * XBUF], &Ws[(p ^ 1) * WBUF], tid);

        const float* xrow  = &Xs[p * XBUF + (wave * 16 + rowM) * XS_STRIDE];
        const float* wrow0 = &Ws[p * WBUF + rowM * WS_STRIDE];          // N = rowM
        const float* wrow1 = &Ws[p * WBUF + (16 + rowM) * WS_STRIDE];   // N = 16+rowM

        #pragma unroll
        for (int kk = 0; kk < BLOCK; kk += 4) {
            const int c = kk + (hiHalf << 1);
            v2f a, b0, b1;
            // A (16x4 f32): lanes0-15 v={K,K+1}, lanes16-31 v={K+2,K+3}, M=rowM
            a.x  = xrow[c];      a.y  = xrow[c + 1];
            // B (4x16 f32): element (K=i, N=o) = w[o][i]; same K striping as A
            b0.x = wrow0[c];     b0.y = wrow0[c + 1];
            b1.x = wrow1[c];     b1.y = wrow1[c + 1];
            acc0 = __builtin_amdgcn_wmma_f32_16x16x4_f32(
                false, a, false, b0, (short)0, acc0, false, false);
            acc1 = __builtin_amdgcn_wmma_f32_16x16x4_f32(
                false, a, false, b1, (short)0, acc1, false, false);
        }
    }

    // D layout (32-bit 16x16): lane<16 -> N=lane, M=j; lane>=16 -> N=lane-16, M=8+j
    float* ybase = y + (size_t)(tokBase + wave * 16) * OUT_F + (size_t)ob * BLOCK;
    #pragma unroll
    for (int j = 0; j < 8; ++j) {
        const int m = j + (hiHalf << 3);
        ybase[(size_t)m * OUT_F + rowM]      = acc0[j];
        ybase[(size_t)m * OUT_F + 16 + rowM] = acc1[j];
    }
}

extern "C" void kernel_launch(void* const* d_in, const int* in_sizes, int n_in,
                              void* d_out, int out_size, void* d_ws, size_t ws_size,
                              hipStream_t stream) {
    const float* x  = (const float*)d_in[0];
    const float* w  = (const float*)d_in[1];
    const int*   oi = (const int*)d_in[2];
    const int*   ii = (const int*)d_in[3];
    float*       y  = (float*)d_out;
    int*         seg = (int*)d_ws;   // 65 ints

    seg_start_kernel<<<1, 128, 0, stream>>>(oi, seg);

    dim3 grid(OBLK, N_TOK / TOK_TILE);   // 64 x 16 workgroups
    bsmm_wmma_kernel<<<grid, 256, 0, stream>>>(x, w, ii, seg, y);
}